// SpikingAuditory_48842368090336
// MI455X (gfx1250) — compile-verified
//
#include <hip/hip_runtime.h>

typedef __attribute__((ext_vector_type(2))) float f32x2;
typedef __attribute__((ext_vector_type(4))) float f32x4;

#define STEPS 10
#define TPB   256

// Issue one 24-byte noise row (3x b64) as CDNA5 async global->LDS DMA.
// lofs: LDS byte offset for this lane's slot; g: global byte address.
__device__ __forceinline__ void async_row24(unsigned lofs, unsigned long long g) {
    unsigned l1 = lofs + 8u, l2 = lofs + 16u;
    unsigned long long g1 = g + 8ull, g2 = g + 16ull;
    asm volatile("global_load_async_to_lds_b64 %0, %1, off" :: "v"(lofs), "v"(g)  : "memory");
    asm volatile("global_load_async_to_lds_b64 %0, %1, off" :: "v"(l1),   "v"(g1) : "memory");
    asm volatile("global_load_async_to_lds_b64 %0, %1, off" :: "v"(l2),   "v"(g2) : "memory");
}

__global__ __launch_bounds__(TPB, 1) void spiking_auditory_kernel(
    const float* __restrict__ pred,
    const float* __restrict__ consp,
    const float* __restrict__ amb,
    const float* __restrict__ onset,
    const float* __restrict__ plow,
    const float* __restrict__ pmid,
    const float* __restrict__ phigh,
    const float* __restrict__ v0,
    const float* __restrict__ u0,
    const float* __restrict__ rate0,
    const float* __restrict__ noise,
    float* __restrict__ out,
    float* __restrict__ startle,
    int B)
{
    // Double-buffered async staging: 2 x 256 lanes x 6 floats = 12 KB LDS.
    __shared__ float stage[2][TPB * 6];

    const int tid = threadIdx.x;
    const int b   = blockIdx.x * TPB + tid;
    if (b >= B) return;

    // LDS byte offsets of this lane's 24B slot in each buffer
    typedef __attribute__((address_space(3))) float lds_f;
    const unsigned lofs0 = (unsigned)(unsigned long long)(lds_f*)&stage[0][tid * 6];
    const unsigned lofs1 = lofs0 + (unsigned)(TPB * 6 * sizeof(float));
    const unsigned lofs[2] = { lofs0, lofs1 };

    // noise [STEPS, B, 6]
    const unsigned long long gbase = (unsigned long long)(noise) + (unsigned long long)b * 24ull;
    const unsigned long long gstep = (unsigned long long)B * 24ull;

    // ---- prologue: start DMA for steps 0 and 1 while we do the band math ----
    async_row24(lofs[0], gbase);
    async_row24(lofs[1], gbase + gstep);

    // ---- frequency-band drives ----
    const float pd = pred[b];
    const float cc = consp[b];
    const float an = amb[b];
    const float so = onset[b];

    float tl   = fmaxf(0.0f, (200.0f - pd) * (1.0f / 200.0f));
    float low  = (pd < 200.0f) ? tl * tl : 0.0f;
    float tm   = fmaxf(0.0f, (150.0f - cc) * (1.0f / 150.0f)) * 0.8f;
    float mid  = (cc < 150.0f) ? tm : 0.0f;
    float high = fminf(1.0f, fmaf(an, 0.3f, so));

    float d_low  = fabsf(low  - plow[b]);
    float d_mid  = fabsf(mid  - pmid[b]);
    float d_high = fabsf(high - phigh[b]);
    float salience = fmaf(0.4f, d_low, fmaf(0.3f, d_mid, 0.3f * d_high));
    bool  st = (so > 0.6f) || ((d_low > 0.4f) && (low > 0.5f));

    // ---- Izhikevich: 140 + I_TONIC baked in (139) ----
    float Ibase[6];
    Ibase[0] = fmaf(low,  12.0f, 139.0f);
    Ibase[1] = fmaf(low,   8.0f, 139.0f);
    Ibase[2] = fmaf(mid,  12.0f, 139.0f);
    Ibase[3] = fmaf(mid,   8.0f, 139.0f);
    Ibase[4] = fmaf(high, 12.0f, 139.0f);
    Ibase[5] = fmaf(high,  8.0f, 139.0f);

    // state loads: [B,6] rows -> 3x b64 coalesced, non-temporal (single use)
    float v[6], u[6], r[6];
    {
        const f32x2* vp = reinterpret_cast<const f32x2*>(v0)    + (size_t)b * 3;
        const f32x2* up = reinterpret_cast<const f32x2*>(u0)    + (size_t)b * 3;
        const f32x2* rp = reinterpret_cast<const f32x2*>(rate0) + (size_t)b * 3;
#pragma unroll
        for (int j = 0; j < 3; ++j) {
            f32x2 tv = __builtin_nontemporal_load(const_cast<f32x2*>(vp + j));
            f32x2 tu = __builtin_nontemporal_load(const_cast<f32x2*>(up + j));
            f32x2 tr = __builtin_nontemporal_load(const_cast<f32x2*>(rp + j));
            v[2*j] = tv.x; v[2*j+1] = tv.y;
            u[2*j] = tu.x; u[2*j+1] = tu.y;
            r[2*j] = tr.x; r[2*j+1] = tr.y;
        }
    }

    // ---- 10-step loop: consume buffer s&1, keep 2 steps of DMA in flight ----
#pragma unroll
    for (int s = 0; s < STEPS; ++s) {
        // Drain exactly step s's 3 async loads (they complete in order);
        // leave step s+1's 3 loads outstanding.
        if (s < STEPS - 1) {
            asm volatile("s_wait_asynccnt 0x3" ::: "memory");
        } else {
            asm volatile("s_wait_asynccnt 0x0" ::: "memory");
        }

        const f32x2* sp = reinterpret_cast<const f32x2*>(&stage[s & 1][tid * 6]);
        f32x2 n0 = sp[0];
        f32x2 n1 = sp[1];
        f32x2 n2 = sp[2];
        float nzv[6] = { n0.x, n0.y, n1.x, n1.y, n2.x, n2.y };

#pragma unroll
        for (int n = 0; n < 6; ++n) {
            // v += 0.04v^2 + 5v + 140 - u + I + 0.3*nz - 1
            //   => v = v*(0.04v + 6) + (Ibase + 0.3*nz - u)
            float Iin = fmaf(nzv[n], 0.3f, Ibase[n]);
            float acc = Iin - u[n];
            float vv  = v[n];
            vv = fmaf(vv, fmaf(0.04f, vv, 6.0f), acc);
            float uu = fmaf(0.02f, fmaf(0.2f, vv, -u[n]), u[n]);
            bool  sp_ = vv >= 30.0f;
            v[n] = sp_ ? -65.0f : vv;
            u[n] = sp_ ? (uu + 8.0f) : uu;
            r[n] = fmaf(0.9f, r[n], sp_ ? 0.1f : 0.0f);
        }

        // Refill the buffer we just consumed with step s+2's row.
        if (s + 2 < STEPS) {
            async_row24(lofs[s & 1], gbase + (unsigned long long)(s + 2) * gstep);
        }
    }

    float rate_mean = (r[0] + r[1] + r[2] + r[3] + r[4] + r[5]) * (1.0f / 6.0f);

    // ---- two-compartment column: prediction == sensory (faithful to ref),
    //      all 4 per-channel units identical (vd0=vs0=0) => 3 scalar ODEs ----
    const float sens[3] = { low, mid, high };
    float pe_mean = 0.0f, prec_mean = 0.0f, free_energy = 0.0f;
#pragma unroll
    for (int c = 0; c < 3; ++c) {
        const float sc = sens[c];
        float vd = 0.0f, vs = 0.0f;
#pragma unroll
        for (int k = 0; k < 8; ++k) {
            vd = fmaf(0.125f, sc - vd, vd);                 // vd += dt*(-vd + p)
            vs = fmaf(0.125f, fmaf(0.5f, vd, sc) - vs, vs); // vs += dt*(-vs + s + 0.5*vd_new)
        }
        float pe   = vs - vd;
        float p2   = pe * pe;
        float prec = 1.0f / (1.0f + p2);
        pe_mean     += fabsf(pe);
        prec_mean   += prec;
        free_energy += fmaf(0.5f * prec, p2, 0.5f * log1pf(p2));
    }
    pe_mean   *= (1.0f / 3.0f);
    prec_mean *= (1.0f / 3.0f);

    // ---- outputs: out[b,8] as 2x b128 NT stores, startle as 0/1 float ----
    f32x4 o0 = { low, mid, high, salience };
    f32x4 o1 = { pe_mean, prec_mean, free_energy, rate_mean };
    f32x4* op = reinterpret_cast<f32x4*>(out) + (size_t)b * 2;
    __builtin_nontemporal_store(o0, op);
    __builtin_nontemporal_store(o1, op + 1);
    startle[b] = st ? 1.0f : 0.0f;
}

extern "C" void kernel_launch(void* const* d_in, const int* in_sizes, int n_in,
                              void* d_out, int out_size, void* d_ws, size_t ws_size,
                              hipStream_t stream) {
    const int B = in_sizes[0];
    const float* pred  = (const float*)d_in[0];
    const float* consp = (const float*)d_in[1];
    const float* amb   = (const float*)d_in[2];
    const float* onset = (const float*)d_in[3];
    const float* plow  = (const float*)d_in[4];
    const float* pmid  = (const float*)d_in[5];
    const float* phigh = (const float*)d_in[6];
    const float* v0    = (const float*)d_in[7];
    const float* u0    = (const float*)d_in[8];
    const float* rate0 = (const float*)d_in[9];
    const float* noise = (const float*)d_in[10];

    float* out     = (float*)d_out;
    float* startle = out + (size_t)B * 8;

    dim3 block(TPB);
    dim3 grid((B + TPB - 1) / TPB);
    hipLaunchKernelGGL(spiking_auditory_kernel, grid, block, 0, stream,
                       pred, consp, amb, onset, plow, pmid, phigh,
                       v0, u0, rate0, noise, out, startle, B);
}